// SoAP_39771397161544
// MI455X (gfx1250) — compile-verified
//
#include <hip/hip_runtime.h>
#include <hip/hip_bf16.h>
#include <math.h>

typedef __attribute__((ext_vector_type(2))) float v2f;
typedef __attribute__((ext_vector_type(8))) float v8f;

#define BATCH   64
#define NPTS    16384
#define DIM     16
#define KSLICES 32
#define ROWS_PER_SLICE (NPTS / KSLICES)   // 512 rows -> 128 WMMA per wave

#define WMMA_F32X4(a, c) \
  __builtin_amdgcn_wmma_f32_16x16x4_f32(false, (a), false, (a), (short)0, (c), false, false)

// ---------------------------------------------------------------------------
// Kernel 0: zero the cov accumulation workspace (64 x 256 floats)
// ---------------------------------------------------------------------------
__global__ void soap_zero_ws(float* __restrict__ ws) {
  int i = blockIdx.x * blockDim.x + threadIdx.x;
  if (i < BATCH * 256) ws[i] = 0.0f;
}

// ---------------------------------------------------------------------------
// Kernel 1: cov[b] = (X^T X) / N via V_WMMA_F32_16X16X4_F32 rank-4 updates.
// One wave per (batch, K-slice). For X^T X the A-tile and B-tile have
// identical per-lane register contents, so one load feeds both operands:
//   lane l, rows k..k+3:  v0 = x[k + 2*(l>=16)][l%16], v1 = next row.
// Each iteration handles 16 rows: 8 independent dword loads (clause together,
// all in flight before the first wait) feeding 4 WMMAs on 2 alternating
// accumulators to avoid back-to-back RAW on the accumulator registers.
// ---------------------------------------------------------------------------
__global__ __launch_bounds__(32) void soap_cov_wmma(const float* __restrict__ x,
                                                    float* __restrict__ cov) {
  const int b     = blockIdx.y;
  const int slice = blockIdx.x;
  const int lane  = threadIdx.x;        // wave32: 0..31
  const int col   = lane & 15;
  const int koff  = (lane >> 4) << 1;   // 0 for lanes 0-15, 2 for lanes 16-31

  const float* __restrict__ xb = x + (size_t)b * (NPTS * DIM);
  const int k0 = slice * ROWS_PER_SLICE;

  v8f c0 = {};
  v8f c1 = {};
  for (int k = k0; k < k0 + ROWS_PER_SLICE; k += 16) {
    // branchless speculative prefetch ~64 rows (4 KB) ahead, clamped in-slab
    int pk = k + 64;
    pk = pk < (NPTS - 16) ? pk : (NPTS - 16);
    __builtin_prefetch(&xb[pk * DIM + col], 0, 1);

    const float* __restrict__ p0 = &xb[(k + koff) * DIM + col];
    v2f a0, a1, a2, a3;
    a0.x = p0[0 * DIM];  a0.y = p0[1 * DIM];
    a1.x = p0[4 * DIM];  a1.y = p0[5 * DIM];
    a2.x = p0[8 * DIM];  a2.y = p0[9 * DIM];
    a3.x = p0[12 * DIM]; a3.y = p0[13 * DIM];

    c0 = WMMA_F32X4(a0, c0);
    c1 = WMMA_F32X4(a1, c1);
    c0 = WMMA_F32X4(a2, c0);
    c1 = WMMA_F32X4(a3, c1);
  }

  // C/D layout: VGPR r -> M = r + 8*(lane>=16), N = lane%16
  float* __restrict__ cb = cov + b * 256;
  const int mb    = (lane >> 4) * 8;
  const float scl = 1.0f / (float)NPTS;
#pragma unroll
  for (int r = 0; r < 8; ++r)
    atomicAdd(&cb[(mb + r) * DIM + col], (c0[r] + c1[r]) * scl);
}

// ---------------------------------------------------------------------------
// Kernel 2: per batch (64 blocks x 256 threads):
//   fp64 parallel cyclic Jacobi eigen of symmetric 16x16 cov (SVD of PSD),
//   m = V diag(log l) V^T, signed power |m|^p, FC (256x256), L2 normalize.
// ---------------------------------------------------------------------------
__global__ __launch_bounds__(256) void soap_eig_fc(const float* __restrict__ cov,
                                                   const float* __restrict__ W,
                                                   const float* __restrict__ bias,
                                                   const float* __restrict__ pw,
                                                   float* __restrict__ out) {
  __shared__ double As[16][16];
  __shared__ double Vs[16][16];
  __shared__ double cs[8], sn[8];
  __shared__ int    pp[8], qq[8];
  __shared__ double ld[16];
  __shared__ float  mf[256];
  __shared__ float  red[256];

  const int b   = blockIdx.x;
  const int tid = threadIdx.x;
  const int i   = tid >> 4;
  const int j   = tid & 15;

  As[i][j] = (double)cov[b * 256 + tid];
  Vs[i][j] = (i == j) ? 1.0 : 0.0;
  __syncthreads();

  // 12 fixed sweeps of cyclic Jacobi; circle-method round robin gives
  // 15 rounds of 8 disjoint pivot pairs covering all 120 pairs.
  for (int sweep = 0; sweep < 12; ++sweep) {
    for (int r = 0; r < 15; ++r) {
      if (tid < 8) {
        const int a0 = (tid == 0) ? 0 : ((tid - 1 + r) % 15) + 1;
        const int jj = 15 - tid;
        const int a1 = ((jj - 1 + r) % 15) + 1;
        const int p = a0 < a1 ? a0 : a1;
        const int q = a0 < a1 ? a1 : a0;
        const double apq = As[p][q];
        double cc = 1.0, ss = 0.0;
        if (fabs(apq) > 1e-300) {
          const double th = (As[q][q] - As[p][p]) / (2.0 * apq);
          const double t  = ((th >= 0.0) ? 1.0 : -1.0) /
                            (fabs(th) + sqrt(th * th + 1.0));
          cc = 1.0 / sqrt(t * t + 1.0);
          ss = t * cc;
        }
        pp[tid] = p; qq[tid] = q; cs[tid] = cc; sn[tid] = ss;
      }
      __syncthreads();
      if (tid < 128) {             // row phase: A <- J^T A
        const int rot = tid >> 4, cidx = tid & 15;
        const int p = pp[rot], q = qq[rot];
        const double cc = cs[rot], ss = sn[rot];
        const double ap = As[p][cidx], aq = As[q][cidx];
        As[p][cidx] = cc * ap - ss * aq;
        As[q][cidx] = ss * ap + cc * aq;
      }
      __syncthreads();
      if (tid < 128) {             // col phase: A <- A J,  V <- V J
        const int rot = tid >> 4, ridx = tid & 15;
        const int p = pp[rot], q = qq[rot];
        const double cc = cs[rot], ss = sn[rot];
        const double ap = As[ridx][p], aq = As[ridx][q];
        As[ridx][p] = cc * ap - ss * aq;
        As[ridx][q] = ss * ap + cc * aq;
        const double vp = Vs[ridx][p], vq = Vs[ridx][q];
        Vs[ridx][p] = cc * vp - ss * vq;
        Vs[ridx][q] = ss * vp + cc * vq;
      }
      __syncthreads();
    }
  }

  if (tid < 16) ld[tid] = log(As[tid][tid]);
  __syncthreads();

  // m[i][j] = sum_k V[i][k] * log(l_k) * V[j][k]; signed power
  const double pe = (double)pw[0];
  double acc = 0.0;
#pragma unroll
  for (int k = 0; k < 16; ++k)
    acc += Vs[i][k] * ld[k] * Vs[j][k];
  const double sg  = (acc > 0.0) ? 1.0 : ((acc < 0.0) ? -1.0 : 0.0);
  mf[tid] = (float)(sg * pow(fabs(acc), pe));
  __syncthreads();

  // FC: f[tid] = bias[tid] + sum_k mf[k] * W[tid][k]
  float facc = bias[tid];
  const float* __restrict__ Wrow = W + (size_t)tid * 256;
#pragma unroll 8
  for (int k = 0; k < 256; ++k)
    facc = fmaf(mf[k], Wrow[k], facc);

  red[tid] = facc * facc;
  __syncthreads();
  for (int s = 128; s > 0; s >>= 1) {
    if (tid < s) red[tid] += red[tid + s];
    __syncthreads();
  }
  const float nrm = fmaxf(sqrtf(red[0]), 1e-12f);
  out[b * 256 + tid] = facc / nrm;
}

// ---------------------------------------------------------------------------
extern "C" void kernel_launch(void* const* d_in, const int* in_sizes, int n_in,
                              void* d_out, int out_size, void* d_ws, size_t ws_size,
                              hipStream_t stream) {
  (void)in_sizes; (void)n_in; (void)out_size; (void)ws_size;
  const float* x    = (const float*)d_in[0];  // [64, 16384, 16]
  const float* W    = (const float*)d_in[1];  // [256, 256]
  const float* bvec = (const float*)d_in[2];  // [256]
  const float* p    = (const float*)d_in[3];  // [1]
  float* out = (float*)d_out;                 // [64, 256]
  float* cov = (float*)d_ws;                  // 64*256 floats

  soap_zero_ws<<<dim3((BATCH * 256 + 255) / 256), dim3(256), 0, stream>>>(cov);
  soap_cov_wmma<<<dim3(KSLICES, BATCH), dim3(32), 0, stream>>>(x, cov);
  soap_eig_fc<<<dim3(BATCH), dim3(256), 0, stream>>>(cov, W, bvec, p, out);
}